// SGC_4449586118805
// MI455X (gfx1250) — compile-verified
//
#include <hip/hip_runtime.h>

// ---------------------------------------------------------------------------
// SGC (2-hop GCN propagation + dense projection) for MI455X / gfx1250.
//   x:[n,64] f32, edge_index:[2,E] i32, edge_weight:[E] f32,
//   kernel:[64,64] f32, bias:[64] f32  ->  out:[n,64] f32
// Propagation is L2-resident scatter/gather (atomic f32 adds);
// final projection uses V_WMMA_F32_16X16X4_F32 (fp32 matrix core path).
// ---------------------------------------------------------------------------

typedef float v2f __attribute__((ext_vector_type(2)));
typedef float v8f __attribute__((ext_vector_type(8)));

#define D     64          // feature dim == units (fixed by reference shapes)
#define FILL  1.0f        // self-loop weight (IMPROVED == False)

// deg[i] = FILL (self-loop contribution to segment_sum)
__global__ __launch_bounds__(256) void k_init_deg(float* __restrict__ deg, int n) {
  int i = blockIdx.x * 256 + threadIdx.x;
  if (i < n) deg[i] = FILL;
}

// deg[row[e]] += w[e]
__global__ __launch_bounds__(256) void k_accum_deg(const int* __restrict__ row,
                                                   const float* __restrict__ w,
                                                   float* __restrict__ deg, int E) {
  int e = blockIdx.x * 256 + threadIdx.x;
  if (e < E) atomicAdd(&deg[row[e]], w[e]);
}

// in-place deg -> dinv = deg>0 ? rsqrt(deg) : 0
__global__ __launch_bounds__(256) void k_dinv(float* __restrict__ deg, int n) {
  int i = blockIdx.x * 256 + threadIdx.x;
  if (i < n) {
    float d = deg[i];
    deg[i] = (d > 0.0f) ? rsqrtf(d) : 0.0f;
  }
}

// norm_w[e] = dinv[row]*w*dinv[col]
__global__ __launch_bounds__(256) void k_normw(const int* __restrict__ row,
                                               const int* __restrict__ col,
                                               const float* __restrict__ w,
                                               const float* __restrict__ dinv,
                                               float* __restrict__ nw, int E) {
  int e = blockIdx.x * 256 + threadIdx.x;
  if (e < E) nw[e] = dinv[row[e]] * w[e] * dinv[col[e]];
}

// h_out[i,:] = (dinv[i]*FILL*dinv[i]) * h_in[i,:]   (self-loop term, also zero-inits)
__global__ __launch_bounds__(256) void k_hop_init(const float* __restrict__ h_in,
                                                  const float* __restrict__ dinv,
                                                  float* __restrict__ h_out, int n) {
  long i = (long)blockIdx.x * 256 + threadIdx.x;       // over n*D elements
  if (i < (long)n * D) {
    int node = (int)(i >> 6);
    float s = dinv[node];
    s = s * s * FILL;
    h_out[i] = h_in[i] * s;
  }
}

// One wave (32 lanes) per edge: gather h_in[col] (2 floats/lane, coalesced 256B),
// scale by wave-uniform norm_w, scatter-add to h_out[row] with f32 atomics
// (no return -> STOREcnt path). Edge scalars are wave-uniform -> scalar loads.
__global__ __launch_bounds__(256) void k_hop_scatter(const int* __restrict__ row,
                                                     const int* __restrict__ col,
                                                     const float* __restrict__ nw,
                                                     const float* __restrict__ h_in,
                                                     float* __restrict__ h_out, int E) {
  int wave = threadIdx.x >> 5;
  int lane = threadIdx.x & 31;
  int e = blockIdx.x * 8 + wave;            // 8 waves per 256-thread block
  if (e >= E) return;
  float s = nw[e];
  long c = col[e];
  long r = row[e];
  float2 v = *(const float2*)(h_in + c * D + 2 * lane);
  float* dst = h_out + r * D + 2 * lane;
  atomicAdd(dst,     v.x * s);
  atomicAdd(dst + 1, v.y * s);
}

// out[n,64] = h[n,64] @ W[64,64] + bias, via V_WMMA_F32_16X16X4_F32.
// Per wave: one 16-row M tile x full N=64 (4 accumulator frags),
// K in steps of 4 -> 64 WMMA ops per wave. W staged in LDS (16 KB).
//
// A-frag (16x4 f32, 2 VGPRs): lanes 0-15 hold A[M=lane][k0], A[M=lane][k0+1];
//                             lanes 16-31 hold A[M=lane-16][k0+2], [k0+3].
// B-frag (4x16 f32, 2 VGPRs): lanes 0-15 hold B[k0][N=lane], B[k0+1][N=lane];
//                             lanes 16-31 hold B[k0+2][N], B[k0+3][N].
// C/D (16x16 f32, 8 VGPRs):   lanes 0-15: VGPRr = C[M=r][N=lane];
//                             lanes 16-31: VGPRr = C[M=8+r][N=lane-16].
__global__ __launch_bounds__(256) void k_gemm_wmma(const float* __restrict__ h,
                                                   const float* __restrict__ W,
                                                   const float* __restrict__ bias,
                                                   float* __restrict__ out, int n) {
  __shared__ float sW[D * D];                // 16 KB of a 320 KB WGP LDS
  for (int i = threadIdx.x; i < D * D; i += 256) sW[i] = W[i];
  __syncthreads();

  int wave = threadIdx.x >> 5;
  int lane = threadIdx.x & 31;
  int m0 = (blockIdx.x * 8 + wave) * 16;     // wave-uniform tile guard:
  if (m0 >= n) return;                       // EXEC stays all-ones for WMMA

  int nlo   = lane & 15;
  int khalf = (lane >> 4) << 1;              // 0 for lanes 0-15, 2 for 16-31

  v8f acc[4] = {};
  const float* arow = h + (long)(m0 + nlo) * D;

#pragma unroll
  for (int k0 = 0; k0 < D; k0 += 4) {
    int kl = k0 + khalf;
    float2 av = *(const float2*)(arow + kl);
    v2f a = {av.x, av.y};
    const float* w0 = sW + kl * D + nlo;     // row kl of W
    const float* w1 = w0 + D;                // row kl+1 of W
#pragma unroll
    for (int t = 0; t < 4; ++t) {
      v2f b = {w0[t * 16], w1[t * 16]};
      acc[t] = __builtin_amdgcn_wmma_f32_16x16x4_f32(
          /*neg_a=*/false, a, /*neg_b=*/false, b,
          /*c_mod=*/(short)0, acc[t], /*reuse_a=*/false, /*reuse_b=*/false);
    }
  }

  int mrow = m0 + ((lane >> 4) << 3);        // +8 row offset for upper lanes
  float* orow = out + (long)mrow * D + nlo;
#pragma unroll
  for (int t = 0; t < 4; ++t) {
    float bv = bias[t * 16 + nlo];
#pragma unroll
    for (int r = 0; r < 8; ++r)
      orow[(long)r * D + t * 16] = acc[t][r] + bv;
  }
}

static inline size_t align256(size_t v) { return (v + 255) & ~(size_t)255; }

extern "C" void kernel_launch(void* const* d_in, const int* in_sizes, int n_in,
                              void* d_out, int out_size, void* d_ws, size_t ws_size,
                              hipStream_t stream) {
  const float* x    = (const float*)d_in[0];
  const int*   ei   = (const int*)d_in[1];
  const float* ew   = (const float*)d_in[2];
  const float* Wk   = (const float*)d_in[3];
  const float* bias = (const float*)d_in[4];

  const int E = in_sizes[2];              // edge count
  const int n = in_sizes[0] / D;          // node count (d_feat = 64)
  const int* row = ei;                    // edge_index[0]
  const int* col = ei + E;                // edge_index[1]

  // Workspace partition (~58 MB): dinv | norm_w | hA | hB
  char* ws = (char*)d_ws;
  size_t o = 0;
  float* dinv = (float*)(ws + o); o += align256((size_t)n * 4);
  float* nw   = (float*)(ws + o); o += align256((size_t)E * 4);
  float* hA   = (float*)(ws + o); o += align256((size_t)n * D * 4);
  float* hB   = (float*)(ws + o);

  const int TPB = 256;
  const int nb_n  = (n + TPB - 1) / TPB;
  const int nb_e  = (E + TPB - 1) / TPB;
  const int nb_nd = (int)(((long)n * D + TPB - 1) / TPB);
  const int nb_sc = (E + 7) / 8;          // one wave per edge

  // Normalization (renorm trick, matches reference exactly)
  k_init_deg <<<nb_n, TPB, 0, stream>>>(dinv, n);
  k_accum_deg<<<nb_e, TPB, 0, stream>>>(row, ew, dinv, E);
  k_dinv     <<<nb_n, TPB, 0, stream>>>(dinv, n);
  k_normw    <<<nb_e, TPB, 0, stream>>>(row, col, ew, dinv, nw, E);

  // Hop 1: x -> hA
  k_hop_init   <<<nb_nd, TPB, 0, stream>>>(x, dinv, hA, n);
  k_hop_scatter<<<nb_sc, TPB, 0, stream>>>(row, col, nw, x, hA, E);

  // Hop 2: hA -> hB
  k_hop_init   <<<nb_nd, TPB, 0, stream>>>(hA, dinv, hB, n);
  k_hop_scatter<<<nb_sc, TPB, 0, stream>>>(row, col, nw, hA, hB, E);

  // Dense projection on the fp32 WMMA path: out = hB @ W + bias
  const int tiles  = (n + 15) / 16;
  const int nb_gem = (tiles + 7) / 8;     // 8 waves (tiles) per block
  k_gemm_wmma<<<nb_gem, TPB, 0, stream>>>(hB, Wk, bias, (float*)d_out, n);
}